// CRF_RNN_15985868275733
// MI455X (gfx1250) — compile-verified
//
#include <hip/hip_runtime.h>

// CRF-RNN mean-field on MI455X (gfx1250).
// Folded formulation: out = O * W,  O = inputs @ W_feat^T  (16384x256),
// W = D*(I + M1 + M1^2 + M1^3) + M1^4  (256x256), M1 = A2^T * D.
// Main GEMM uses V_WMMA_F32_16X16X4_F32 with LDS-resident, fragment-swizzled W.

typedef __attribute__((ext_vector_type(2))) float v2f;
typedef __attribute__((ext_vector_type(4))) float v4f;
typedef __attribute__((ext_vector_type(8))) float v8f;

#define T_DIM 16384
#define N_DIM 256
#define M_DIM 8
#define K_DIM 16
#define TT    32          // t-rows per block in main kernel

// ---------------------------------------------------------------------------
// Kernel 1: A2[i][j] = 2*sum_k kernels[i,j,k]*W_lin[k];  rd[i] = 1/denom[i]
// grid = N (i), block = N (j)
// ---------------------------------------------------------------------------
__global__ void prep_a2(const float* __restrict__ kernels,
                        const float* __restrict__ W_feat,
                        const float* __restrict__ W_lin,
                        float* __restrict__ tmpA,
                        float* __restrict__ rd) {
    __shared__ float red[N_DIM];
    const int i = blockIdx.x;
    const int j = threadIdx.x;
    const float* kr = kernels + (i * N_DIM + j) * K_DIM;
    float s = 0.f;
#pragma unroll
    for (int k = 0; k < K_DIM; ++k) s = fmaf(kr[k], W_lin[k], s);
    const float a2 = 2.0f * s;
    tmpA[i * N_DIM + j] = a2;
    red[j] = a2;
    __syncthreads();
    for (int off = N_DIM / 2; off > 0; off >>= 1) {
        if (j < off) red[j] += red[j + off];
        __syncthreads();
    }
    if (j == 0) {
        float fw = 0.f;
#pragma unroll
        for (int m = 0; m < M_DIM; ++m) fw += W_feat[m];
        rd[i] = 1.0f / (fw + red[0]);
    }
}

// ---------------------------------------------------------------------------
// Kernel 2: M1[j][i] = A2[i][j]*rd[i];  Q = I + M1;  P = M1
// grid = N (j), block = N (i)
// ---------------------------------------------------------------------------
__global__ void build_m1(const float* __restrict__ tmpA,
                         const float* __restrict__ rd,
                         float* __restrict__ M1,
                         float* __restrict__ Q,
                         float* __restrict__ P) {
    const int j = blockIdx.x;
    const int i = threadIdx.x;
    const float v = tmpA[i * N_DIM + j] * rd[i];
    M1[j * N_DIM + i] = v;
    P [j * N_DIM + i] = v;
    Q [j * N_DIM + i] = v + (i == j ? 1.0f : 0.0f);
}

// ---------------------------------------------------------------------------
// Kernel 3: 256x256x256 matmul C = A*B.
// mode 0: also Q += C.   mode 1: Wm[r][c] = rd[r]*Q[r][c] + (A*B)[r][c]
// grid = N (row r), block = N (col c)
// ---------------------------------------------------------------------------
__global__ void mm256(const float* __restrict__ A,
                      const float* __restrict__ B,
                      float* __restrict__ C,
                      float* __restrict__ Q,
                      const float* __restrict__ rd,
                      float* __restrict__ Wm,
                      int mode) {
    const int r = blockIdx.x;
    const int c = threadIdx.x;
    float acc = 0.f;
    for (int j = 0; j < N_DIM; ++j)
        acc = fmaf(A[r * N_DIM + j], B[j * N_DIM + c], acc);
    if (mode == 0) {
        C[r * N_DIM + c] = acc;
        Q[r * N_DIM + c] += acc;
    } else {
        Wm[r * N_DIM + c] = rd[r] * Q[r * N_DIM + c] + acc;
    }
}

// ---------------------------------------------------------------------------
// Main kernel: out[tb..tb+31, 0..255] = O_tile * W   (WMMA f32 16x16x4)
//
// LDS layout (dynamic, 288 KB total):
//   sW: 65536 floats — W swizzled to B-fragment order:
//       frag(ct,kk) = 64 floats; lane l<16 holds {W[4kk][16ct+l], W[4kk+1][16ct+l]}
//                     lane l>=16 holds {W[4kk+2][16ct+l-16], W[4kk+3][16ct+l-16]}
//       addr = ((ct*64 + kk)*32 + lane)*2 + elem
//   sO: 8192 floats — O tile swizzled to A-fragment order:
//       frag(s,kk): lane l<16 holds {O[16s+l][4kk], O[16s+l][4kk+1]}
//                   lane l>=16 holds {O[16s+l-16][4kk+2], O[16s+l-16][4kk+3]}
//       addr = ((s*64 + kk)*32 + lane)*2 + elem
// Each of 8 waves: stripe s = wave/4, colgroup g = wave%4, owns col-tiles
// c = 4g..4g+3 (4 x v8f accumulators), 64 k-steps of wmma_f32_16x16x4_f32.
// ---------------------------------------------------------------------------
__launch_bounds__(256, 1)
__global__ void crf_main(const float* __restrict__ inputs,
                         const float* __restrict__ W_feat,
                         const float* __restrict__ Wmat,
                         float* __restrict__ out) {
    extern __shared__ float lds[];
    float* sW = lds;            // 65536 floats
    float* sO = lds + 65536;    // 8192 floats

    const int tid = threadIdx.x;
    const int tb  = blockIdx.x * TT;

    // ---- stage W into LDS (swizzled) : 16384 float4 chunks / 256 threads ----
    for (int idx = tid; idx < (N_DIM * N_DIM) / 4; idx += 256) {
        const int j  = idx >> 6;            // W row 0..255
        const int i4 = (idx & 63) << 2;     // W col base (mult of 4)
        const v4f w = *(const v4f*)(Wmat + j * N_DIM + i4);
        const int kk = j >> 2, r = j & 3;
        const int laneBase = (r < 2) ? 0 : 16;
        const int elem = r & 1;
#pragma unroll
        for (int q = 0; q < 4; ++q) {
            const int i  = i4 + q;
            const int ct = i >> 4, li = i & 15;
            sW[((ct * 64 + kk) << 6) + ((laneBase + li) << 1) + elem] = w[q];
        }
    }

    // ---- compute O tile (32 x 256) from inputs, store swizzled ----
    float wf[M_DIM];
#pragma unroll
    for (int m = 0; m < M_DIM; ++m) wf[m] = W_feat[m];

#pragma unroll
    for (int it = 0; it < (TT * N_DIM) / 256; ++it) {   // 32 iterations
        const int idx = it * 256 + tid;
        const int tl  = idx >> 8;       // local t row 0..31
        const int j   = idx & 255;      // region
        const float* ip = inputs + ((size_t)(tb + tl) * N_DIM + j) * M_DIM;
        const v4f a = *(const v4f*)ip;
        const v4f b = *(const v4f*)(ip + 4);
        float o = a.x * wf[0];
        o = fmaf(a.y, wf[1], o); o = fmaf(a.z, wf[2], o); o = fmaf(a.w, wf[3], o);
        o = fmaf(b.x, wf[4], o); o = fmaf(b.y, wf[5], o);
        o = fmaf(b.z, wf[6], o); o = fmaf(b.w, wf[7], o);
        const int s = tl >> 4, tr = tl & 15;
        const int kk = j >> 2, r = j & 3;
        const int lane = ((r < 2) ? 0 : 16) + tr;
        const int elem = r & 1;
        sO[((s * 64 + kk) << 6) + (lane << 1) + elem] = o;
    }
    __syncthreads();

    // ---- WMMA GEMM: D(16x16) += A(16x4) * B(4x16), 64 k-steps ----
    const int wave = tid >> 5;
    const int lane = tid & 31;
    const int s = wave >> 2;        // t stripe 0..1
    const int g = wave & 3;         // column group 0..3 -> tiles 4g..4g+3

    v8f acc0 = {}, acc1 = {}, acc2 = {}, acc3 = {};

    const float* aBase = sO + ((s * 64) << 6) + (lane << 1);
    const float* bB0 = sW + (((g * 4 + 0) * 64) << 6) + (lane << 1);
    const float* bB1 = sW + (((g * 4 + 1) * 64) << 6) + (lane << 1);
    const float* bB2 = sW + (((g * 4 + 2) * 64) << 6) + (lane << 1);
    const float* bB3 = sW + (((g * 4 + 3) * 64) << 6) + (lane << 1);

#pragma unroll 8
    for (int kk = 0; kk < 64; ++kk) {
        const v2f a  = *(const v2f*)(aBase + (kk << 6));
        const v2f b0 = *(const v2f*)(bB0 + (kk << 6));
        const v2f b1 = *(const v2f*)(bB1 + (kk << 6));
        const v2f b2 = *(const v2f*)(bB2 + (kk << 6));
        const v2f b3 = *(const v2f*)(bB3 + (kk << 6));
        acc0 = __builtin_amdgcn_wmma_f32_16x16x4_f32(false, a, false, b0,
                                                     (short)0, acc0, false, false);
        acc1 = __builtin_amdgcn_wmma_f32_16x16x4_f32(false, a, false, b1,
                                                     (short)0, acc1, false, false);
        acc2 = __builtin_amdgcn_wmma_f32_16x16x4_f32(false, a, false, b2,
                                                     (short)0, acc2, false, false);
        acc3 = __builtin_amdgcn_wmma_f32_16x16x4_f32(false, a, false, b3,
                                                     (short)0, acc3, false, false);
    }

    // ---- store D tiles: VGPR v -> rows v (lanes 0-15) and v+8 (lanes 16-31) ----
    const int row0 = tb + s * 16;
    const int mAdd = (lane < 16) ? 0 : 8;
    const int ncol = lane & 15;
#pragma unroll
    for (int v = 0; v < 8; ++v) {
        const int row = row0 + v + mAdd;
        float* op = out + (size_t)row * N_DIM + ncol;
        op[(g * 4 + 0) * 16] = acc0[v];
        op[(g * 4 + 1) * 16] = acc1[v];
        op[(g * 4 + 2) * 16] = acc2[v];
        op[(g * 4 + 3) * 16] = acc3[v];
    }
}

// ---------------------------------------------------------------------------
extern "C" void kernel_launch(void* const* d_in, const int* in_sizes, int n_in,
                              void* d_out, int out_size, void* d_ws, size_t ws_size,
                              hipStream_t stream) {
    const float* inputs  = (const float*)d_in[0];   // (T, N, M)
    const float* kernels = (const float*)d_in[1];   // (N, N, K)
    const float* W_feat  = (const float*)d_in[2];   // (1, M)
    const float* W_lin   = (const float*)d_in[3];   // (1, K)
    float* out = (float*)d_out;                     // (T, N, 1)

    float* ws = (float*)d_ws;
    const int NN = N_DIM * N_DIM;
    float* tmpA = ws;            // 64K floats
    float* M1   = ws + 1 * NN;
    float* Pa   = ws + 2 * NN;
    float* Pb   = ws + 3 * NN;
    float* Q    = ws + 4 * NN;
    float* Wm   = ws + 5 * NN;
    float* rd   = ws + 6 * NN;   // 256 floats

    prep_a2 <<<N_DIM, N_DIM, 0, stream>>>(kernels, W_feat, W_lin, tmpA, rd);
    build_m1<<<N_DIM, N_DIM, 0, stream>>>(tmpA, rd, M1, Q, Pa);
    // M1^2, Q += ; M1^3, Q += ; final: Wm = rd[r]*Q + M1^4
    mm256<<<N_DIM, N_DIM, 0, stream>>>(Pa, M1, Pb, Q, rd, Wm, 0);
    mm256<<<N_DIM, N_DIM, 0, stream>>>(Pb, M1, Pa, Q, rd, Wm, 0);
    mm256<<<N_DIM, N_DIM, 0, stream>>>(Pa, M1, Pb, Q, rd, Wm, 1);

    const size_t shmem = (size_t)(65536 + 8192) * sizeof(float);  // 288 KB
    crf_main<<<T_DIM / TT, 256, shmem, stream>>>(inputs, W_feat, Wm, out);
}